// AlignmentHub_70523363000367
// MI455X (gfx1250) — compile-verified
//
#include <hip/hip_runtime.h>
#include <stdint.h>

// Problem geometry (fixed by the reference: B=4096 rows, N=8192 cols).
#define N_COLS   8192
#define NTHREADS 256
#define NWAVES   (NTHREADS / 32)
#define PER_T    (N_COLS / NTHREADS)   // 32 elements per thread
#define PER_T4   (PER_T / 4)           // 8 float4 per thread
#define K_TOP    10

#define W_GAP    0.45f
#define W_ENT    0.35f
#define W_CONS   0.20f
#define A_MIN    0.05f
#define A_MAX    0.95f
#define INV_TEMP (1.0f / 0.07f)
#define INV_LOG_N 0.1109770f           // 1 / ln(8192)
#define NEG_BIG  (-3.402823466e38f)
#define POS_BIG  ( 3.402823466e38f)

// ---------------------------------------------------------------------------
// CDNA5 Tensor Data Mover: async DMA of one 32KB row (global -> LDS).
// Descriptor layout per CDNA5 ISA ch.8 (D# groups). clang-23 6-arg builtin.
// ---------------------------------------------------------------------------
#if defined(__HIP_DEVICE_COMPILE__) && defined(__gfx1250__) && __has_builtin(__builtin_amdgcn_tensor_load_to_lds)
#define HAVE_TDM 1
#else
#define HAVE_TDM 0
#endif

#if HAVE_TDM
typedef uint32_t tdm_v4u __attribute__((ext_vector_type(4)));
typedef int32_t  tdm_v8i __attribute__((ext_vector_type(8)));
typedef int32_t  tdm_v4i __attribute__((ext_vector_type(4)));

__device__ __forceinline__ void tdm_load_row(const float* gsrc, float* lds_dst, int n) {
  uint64_t ga = (uint64_t)(uintptr_t)gsrc;
  // D# group 0: count=1 valid user descriptor; lds_addr; 57-bit global addr; type=2 ("image")
  tdm_v4u g0;
  g0.x = 1u;
  g0.y = (uint32_t)(uintptr_t)lds_dst;                        // flat LDS ptr low 32 bits == LDS byte offset
  g0.z = (uint32_t)ga;                                        // global_addr[31:0]
  g0.w = ((uint32_t)(ga >> 32) & 0x01FFFFFFu) | (2u << 30);   // global_addr[56:32] | type=2
  // D# group 1: data_size=4B (code 2), 1-D tile: tensor_dim0 = tile_dim0 = n
  tdm_v8i g1;
  g1[0] = 0x00020000;                                  // [17:16]=2 (4B); mask/pad/iterate/barrier = 0
  g1[1] = (int)(((uint32_t)n & 0xFFFFu) << 16);        // tensor_dim0[15:0] at bits[63:48]
  g1[2] = (int)((((uint32_t)n >> 16) & 0xFFFFu) | (1u << 16)); // tensor_dim0[31:16], tensor_dim1=1
  g1[3] = (int)(((uint32_t)n & 0xFFFFu) << 16);        // tile_dim0 at bits[127:112]
  g1[4] = 1;                                           // tile_dim1=1, tile_dim2=0
  g1[5] = n;                                           // tensor_dim0_stride[31:0]
  g1[6] = (int)(((uint32_t)n & 0xFFFFu) << 16);        // tensor_dim1_stride[15:0] (unused, Y=0 only)
  g1[7] = 0;
  tdm_v4i gz4 = {0, 0, 0, 0};                          // groups 2/3 unused (<=2D tensor)
  tdm_v8i gz8 = {0, 0, 0, 0, 0, 0, 0, 0};              // extra descriptor words unused
  __builtin_amdgcn_tensor_load_to_lds(g0, g1, gz4, gz4, gz8, 0);
}
#endif

// ---------------------------------------------------------------------------
// wave32 shfl-butterfly block reductions (2 barriers each, value broadcast).
// ---------------------------------------------------------------------------
__device__ __forceinline__ void block_minmax2(float& vmin, float& m1, float& m2,
                                              float* wmin, float* w1, float* w2, int tid) {
  #pragma unroll
  for (int off = 16; off > 0; off >>= 1) {
    vmin = fminf(vmin, __shfl_xor(vmin, off, 32));
    float o1 = __shfl_xor(m1, off, 32);
    float o2 = __shfl_xor(m2, off, 32);
    float n1 = fmaxf(m1, o1);
    float n2 = fmaxf(fminf(m1, o1), fmaxf(m2, o2));
    m1 = n1; m2 = n2;
  }
  const int wid = tid >> 5, lane = tid & 31;
  if (lane == 0) { wmin[wid] = vmin; w1[wid] = m1; w2[wid] = m2; }
  __syncthreads();
  vmin = wmin[0]; m1 = w1[0]; m2 = w2[0];
  #pragma unroll
  for (int j = 1; j < NWAVES; ++j) {
    vmin = fminf(vmin, wmin[j]);
    float o1 = w1[j], o2 = w2[j];
    float n1 = fmaxf(m1, o1);
    float n2 = fmaxf(fminf(m1, o1), fmaxf(m2, o2));
    m1 = n1; m2 = n2;
  }
  __syncthreads();
}

__device__ __forceinline__ void block_sum2(float& a, float& b,
                                           float* wa, float* wb, int tid) {
  #pragma unroll
  for (int off = 16; off > 0; off >>= 1) {
    a += __shfl_xor(a, off, 32);
    b += __shfl_xor(b, off, 32);
  }
  const int wid = tid >> 5, lane = tid & 31;
  if (lane == 0) { wa[wid] = a; wb[wid] = b; }
  __syncthreads();
  a = wa[0]; b = wb[0];
  #pragma unroll
  for (int j = 1; j < NWAVES; ++j) { a += wa[j]; b += wb[j]; }
  __syncthreads();
}

__device__ __forceinline__ void block_argmax(float& bv, int& bi,
                                             float* wv, int* wi, int tid) {
  #pragma unroll
  for (int off = 16; off > 0; off >>= 1) {
    float ov = __shfl_xor(bv, off, 32);
    int   oi = __shfl_xor(bi, off, 32);
    if (ov > bv || (ov == bv && oi < bi)) { bv = ov; bi = oi; }
  }
  const int wid = tid >> 5, lane = tid & 31;
  if (lane == 0) { wv[wid] = bv; wi[wid] = bi; }
  __syncthreads();
  bv = wv[0]; bi = wi[0];
  #pragma unroll
  for (int j = 1; j < NWAVES; ++j) {
    float ov = wv[j]; int oi = wi[j];
    if (ov > bv || (ov == bv && oi < bi)) { bv = ov; bi = oi; }
  }
  __syncthreads();
}

// ---------------------------------------------------------------------------
// One block per row. Both rows TDM'd into LDS up front (double buffer) so the
// g-row DMA overlaps s-row processing; final fused write reads only LDS.
// ---------------------------------------------------------------------------
extern __shared__ __align__(16) float dyn_lds[];   // [2 * N_COLS] row buffers

__global__ __launch_bounds__(NTHREADS)
void alignment_hub_fuse(const float* __restrict__ S, const float* __restrict__ G,
                        float* __restrict__ out) {
  __shared__ float wA[NWAVES];
  __shared__ float wB[NWAVES];
  __shared__ float wC[NWAVES];
  __shared__ int   wI[NWAVES];
  __shared__ int   topIdx[2][K_TOP];
  __shared__ float savVal[2][K_TOP];
  __shared__ float sh_alpha;

  const int row = blockIdx.x;
  const int tid = threadIdx.x;

  float* shs = dyn_lds;
  float* shg = dyn_lds + N_COLS;
  float* rowBuf[2] = { shs, shg };
  const float* srcPtr[2] = { S + (size_t)row * N_COLS, G + (size_t)row * N_COLS };
  float rp_min[2], rp_inv[2], rp_gap[2], rp_ent[2];

  // ---- Kick off both async row DMAs (in-order completion per wave) --------
#if HAVE_TDM
  if (tid == 0) {
    tdm_load_row(srcPtr[0], shs, N_COLS);
    tdm_load_row(srcPtr[1], shg, N_COLS);
  }
#else
  {
    const float4* s4 = (const float4*)srcPtr[0];
    const float4* g4 = (const float4*)srcPtr[1];
    float4* ds = (float4*)shs;
    float4* dg = (float4*)shg;
    #pragma unroll
    for (int i = 0; i < PER_T4; ++i) {
      int j = i * NTHREADS + tid;
      ds[j] = s4[j];
      dg[j] = g4[j];
    }
  }
#endif

  #pragma unroll
  for (int src = 0; src < 2; ++src) {
    float* sh = rowBuf[src];

    // ---- Wait for this row's DMA (s-row: allow g-row still in flight) ----
#if HAVE_TDM
    if (tid == 0) {
      if (src == 0) __builtin_amdgcn_s_wait_tensorcnt(1);
      else          __builtin_amdgcn_s_wait_tensorcnt(0);
    }
#endif
    __syncthreads();

    // ---- Pass 1: min, max (m1), second max (m2) --------------------------
    float lmin = POS_BIG, m1 = NEG_BIG, m2 = NEG_BIG;
    {
      const float4* s4 = (const float4*)sh;
      #pragma unroll
      for (int i = 0; i < PER_T4; ++i) {
        float4 v4 = s4[i * NTHREADS + tid];
        float vv[4] = { v4.x, v4.y, v4.z, v4.w };
        #pragma unroll
        for (int c = 0; c < 4; ++c) {
          float v = vv[c];
          lmin = fminf(lmin, v);
          if (v > m1) { m2 = m1; m1 = v; }
          else if (v > m2) { m2 = v; }
        }
      }
    }
    block_minmax2(lmin, m1, m2, wA, wB, wC, tid);   // broadcast to all threads

    float rng  = m1 - lmin + 1e-6f;
    float inv  = 1.0f / rng;
    float xmax = (m1 - lmin) * inv;            // normalized top value (~1)
    float x2   = (m2 - lmin) * inv;            // normalized second value

    // ---- Pass 2: stabilized softmax sums ---------------------------------
    // e_i = exp((x_i - xmax)/T);  se = sum e_i;  sxe = sum e_i * x_i / T
    float se = 0.0f, sxe = 0.0f;
    {
      const float4* s4 = (const float4*)sh;
      #pragma unroll
      for (int i = 0; i < PER_T4; ++i) {
        float4 v4 = s4[i * NTHREADS + tid];
        float vv[4] = { v4.x, v4.y, v4.z, v4.w };
        #pragma unroll
        for (int c = 0; c < 4; ++c) {
          float x  = (vv[c] - lmin) * inv;
          float xt = x * INV_TEMP;
          float e  = __expf(xt - xmax * INV_TEMP);
          se  += e;
          sxe += e * xt;
        }
      }
    }
    block_sum2(se, sxe, wA, wB, tid);

    // gap = p1 - p2 = (1 - exp((x2-xmax)/T)) / se
    // ent = (log se + xmax/T - sxe/se) / log(N)
    rp_min[src] = lmin;
    rp_inv[src] = inv;
    rp_gap[src] = (1.0f - __expf((x2 - xmax) * INV_TEMP)) / se;
    rp_ent[src] = (__logf(se) + xmax * INV_TEMP - sxe / se) * INV_LOG_N;

    // ---- Top-10 indices: 10 arg-max reductions with knockout + save ------
    for (int k = 0; k < K_TOP; ++k) {
      float bv = NEG_BIG; int bi = 0;
      #pragma unroll
      for (int i = 0; i < PER_T; ++i) {
        int idx = i * NTHREADS + tid;
        float v = sh[idx];
        if (v > bv) { bv = v; bi = idx; }
      }
      block_argmax(bv, bi, wA, wI, tid);
      if (tid == 0) {
        topIdx[src][k] = bi;
        savVal[src][k] = bv;
        sh[bi] = NEG_BIG;            // knock out for next iteration
      }
      __syncthreads();
    }
    // restore knocked-out values so the final pass can read LDS only
    if (tid < K_TOP) sh[topIdx[src][tid]] = savVal[src][tid];
    __syncthreads();
  }

  // ---- Overlap + alpha ----------------------------------------------------
  if (tid == 0) {
    int hits = 0;
    #pragma unroll
    for (int i = 0; i < K_TOP; ++i) {
      int si = topIdx[0][i];
      #pragma unroll
      for (int j = 0; j < K_TOP; ++j) hits += (si == topIdx[1][j]) ? 1 : 0;
    }
    float ov = (float)hits * (1.0f / (float)K_TOP);
    float conf_s = W_GAP * rp_gap[0] + W_ENT * (1.0f - rp_ent[0]) + W_CONS * ov;
    float conf_g = W_GAP * rp_gap[1] + W_ENT * (1.0f - rp_ent[1]) + W_CONS * ov;
    float a = conf_s / (conf_s + conf_g + 1e-6f);
    sh_alpha = fminf(fmaxf(a, A_MIN), A_MAX);
  }
  __syncthreads();

  // ---- Final pass: fused = a*s_fuse + (1-a)*g_fuse, LDS -> global b128 ----
  const float a  = sh_alpha;
  const float b  = 1.0f - a;
  const float smin = rp_min[0], sinv = rp_inv[0];
  const float gmin = rp_min[1], ginv = rp_inv[1];
  const float4* S4 = (const float4*)shs;
  const float4* G4 = (const float4*)shg;
  float4* O4 = (float4*)(out + (size_t)row * N_COLS);
  #pragma unroll
  for (int i = 0; i < PER_T4; ++i) {
    int j = i * NTHREADS + tid;
    float4 sv = S4[j];
    float4 gv = G4[j];
    float4 o;
    o.x = a * ((sv.x - smin) * sinv) + b * ((gv.x - gmin) * ginv);
    o.y = a * ((sv.y - smin) * sinv) + b * ((gv.y - gmin) * ginv);
    o.z = a * ((sv.z - smin) * sinv) + b * ((gv.z - gmin) * ginv);
    o.w = a * ((sv.w - smin) * sinv) + b * ((gv.w - gmin) * ginv);
    O4[j] = o;
  }
}

// ---------------------------------------------------------------------------
extern "C" void kernel_launch(void* const* d_in, const int* in_sizes, int n_in,
                              void* d_out, int out_size, void* d_ws, size_t ws_size,
                              hipStream_t stream) {
  (void)n_in; (void)out_size; (void)d_ws; (void)ws_size;
  const float* S = (const float*)d_in[0];
  const float* G = (const float*)d_in[1];
  float* out = (float*)d_out;
  const int rows = in_sizes[0] / N_COLS;   // 4096
  const size_t shbytes = (size_t)2 * N_COLS * sizeof(float);  // 64 KB row buffers
  alignment_hub_fuse<<<dim3(rows), dim3(NTHREADS), shbytes, stream>>>(S, G, out);
}